// LlamaMorExpertRouter_30477087933212
// MI455X (gfx1250) — compile-verified
//
#include <hip/hip_runtime.h>
#include <hip/hip_bf16.h>

typedef __attribute__((ext_vector_type(2))) float v2f;
typedef __attribute__((ext_vector_type(8))) float v8f;

#define B_SZ 4
#define S_SZ 4096
#define H_SZ 2048
#define K_SEL 2048
#define ALPHA 0.1f

// ---------------------------------------------------------------------------
// Kernel 1: router projection via V_WMMA_F32_16X16X4_F32.
// Each wave computes logits for 16 tokens. A = x[16 tokens x 4 k] (fp32),
// B = w[k0..k0+3] broadcast to all 16 columns -> every column of D equals the
// dot product, so lanes 0 / 16 hold 8 token results each in the v8f acc.
// 8 waves per 256-thread block -> 128 tokens/block -> 128 blocks.
// ---------------------------------------------------------------------------
__global__ __launch_bounds__(256)
void mor_router_wmma(const float* __restrict__ x,
                     const float* __restrict__ w,
                     float* __restrict__ probs) {
    __shared__ float wlds[H_SZ];
    for (int i = threadIdx.x; i < H_SZ; i += 256) wlds[i] = w[i];
    __syncthreads();

    const int wave = threadIdx.x >> 5;            // 0..7
    const int lane = threadIdx.x & 31;
    const int tok_base = (blockIdx.x * 8 + wave) * 16;
    const int m  = lane & 15;                     // token row within tile
    const int kp = (lane >> 4) * 2;               // k-pair: 0 (lanes 0-15) / 2 (16-31)

    const float* __restrict__ xrow = x + (size_t)(tok_base + m) * H_SZ;

    v8f acc = {};
    for (int k0 = 0; k0 < H_SZ; k0 += 4) {
        v2f a, bm;
        a.x  = xrow[k0 + kp];
        a.y  = xrow[k0 + kp + 1];
        bm.x = wlds[k0 + kp];
        bm.y = wlds[k0 + kp + 1];
        acc = __builtin_amdgcn_wmma_f32_16x16x4_f32(
            /*neg_a=*/false, a, /*neg_b=*/false, bm,
            /*c_mod=*/(short)0, acc, /*reuse_a=*/false, /*reuse_b=*/false);
    }

    // lanes 0-15: acc[r] = logit[tok_base+r]; lanes 16-31: acc[r] = logit[tok_base+8+r]
    if (lane == 0 || lane == 16) {
        const int base = tok_base + ((lane >> 4) << 3);
#pragma unroll
        for (int r = 0; r < 8; ++r) {
            float logit = acc[r];
            probs[base + r] = ALPHA / (1.0f + __expf(-logit));
        }
    }
}

// ---------------------------------------------------------------------------
// Kernel 2: per-batch-row exact top-k (k = 2048 of 4096) + ascending-index
// compaction. probs are all positive -> uint bit pattern is order-preserving.
// Threshold T = k-th largest found by 32-step bitwise bisection; ties at T
// resolved by earliest index (matches top_k then sort-ascending semantics).
// One 1024-thread block per batch row, 4 elements per thread.
// ---------------------------------------------------------------------------
__global__ __launch_bounds__(1024)
void mor_select(const float* __restrict__ probs,
                float* __restrict__ sel_f,      // [B][K] (float indices, d_out)
                float* __restrict__ targets,    // [B][S] (d_out)
                int*   __restrict__ sel_i,      // [B][K] (ws)
                float* __restrict__ w_top) {    // [B][K] (ws)
    const int b = blockIdx.x;
    const int t = threadIdx.x;                  // 0..1023

    __shared__ float    p[S_SZ];
    __shared__ unsigned red[1024];
    __shared__ unsigned sGT[1024];
    __shared__ unsigned sEQ[1024];

    const float* __restrict__ pb = probs + b * S_SZ;
    for (int i = t; i < S_SZ; i += 1024) p[i] = pb[i];
    __syncthreads();

    unsigned pu[4];
#pragma unroll
    for (int j = 0; j < 4; ++j) pu[j] = __float_as_uint(p[4 * t + j]);

    // --- bitwise radix bisection: max T with count(>=T) >= K_SEL ---
    unsigned T = 0u;
    for (int bit = 31; bit >= 0; --bit) {
        const unsigned cand = T | (1u << bit);
        unsigned c = 0;
#pragma unroll
        for (int j = 0; j < 4; ++j) c += (pu[j] >= cand) ? 1u : 0u;
        red[t] = c;
        __syncthreads();
        for (int off = 512; off > 0; off >>= 1) {
            if (t < off) red[t] += red[t + off];
            __syncthreads();
        }
        const unsigned total = red[0];
        __syncthreads();
        if (total >= K_SEL) T = cand;
    }

    // --- per-thread (GT, EQ) counts, block-wide inclusive scan ---
    unsigned lgt = 0, leq = 0;
#pragma unroll
    for (int j = 0; j < 4; ++j) {
        lgt += (pu[j] > T) ? 1u : 0u;
        leq += (pu[j] == T) ? 1u : 0u;
    }
    sGT[t] = lgt; sEQ[t] = leq;
    __syncthreads();
    for (int off = 1; off < 1024; off <<= 1) {
        const unsigned ag = (t >= off) ? sGT[t - off] : 0u;
        const unsigned ae = (t >= off) ? sEQ[t - off] : 0u;
        __syncthreads();
        sGT[t] += ag; sEQ[t] += ae;
        __syncthreads();
    }
    unsigned gpre = (t > 0) ? sGT[t - 1] : 0u;   // exclusive prefixes
    unsigned epre = (t > 0) ? sEQ[t - 1] : 0u;
    const unsigned totGT = sGT[1023];
    const unsigned need  = K_SEL - totGT;        // EQ elements to take (earliest idx)

    // --- ordered compaction + targets ---
#pragma unroll
    for (int j = 0; j < 4; ++j) {
        const int s = 4 * t + j;
        const unsigned u = pu[j];
        const bool isgt = (u > T);
        const bool iseq = (u == T);
        const bool sel  = isgt || (iseq && (epre < need));
        if (sel) {
            const unsigned pos = gpre + (epre < need ? epre : need);
            sel_f[b * K_SEL + pos] = (float)s;
            sel_i[b * K_SEL + pos] = s;
            w_top[b * K_SEL + pos] = __uint_as_float(u);
        }
        targets[b * S_SZ + s] = sel ? 1.0f : 0.0f;
        gpre += isgt ? 1u : 0u;
        epre += iseq ? 1u : 0u;
    }
}

// ---------------------------------------------------------------------------
// Kernel 3: gather + scale. One block per selected row; 2048 floats moved as
// float4 (global_load_b128 / global_store_b128). x fits in 192MB L2, so this
// second pass over half of x should be L2-resident.
// ---------------------------------------------------------------------------
__global__ __launch_bounds__(256)
void mor_gather(const float* __restrict__ x,
                const int*   __restrict__ sel_i,
                const float* __restrict__ w_top,
                float* __restrict__ out) {
    const int row = blockIdx.x;                 // 0 .. B*K-1
    const int b   = row >> 11;                  // row / 2048
    const int s   = sel_i[row];
    const float w = w_top[row];

    const float4* __restrict__ src =
        (const float4*)(x + ((size_t)b * S_SZ + (size_t)s) * H_SZ);
    float4* __restrict__ dst = (float4*)(out + (size_t)row * H_SZ);

    __builtin_prefetch(src + threadIdx.x, 0, 3);   // global_prefetch_b8
#pragma unroll
    for (int i = threadIdx.x; i < H_SZ / 4; i += 256) {
        float4 v = src[i];
        v.x *= w; v.y *= w; v.z *= w; v.w *= w;
        dst[i] = v;
    }
}

extern "C" void kernel_launch(void* const* d_in, const int* in_sizes, int n_in,
                              void* d_out, int out_size, void* d_ws, size_t ws_size,
                              hipStream_t stream) {
    const float* x = (const float*)d_in[0];          // [4,4096,2048] f32
    const float* w = (const float*)d_in[1];          // [1,2048] f32
    // d_in[2] = top_k (device scalar) -- compile-time constant 2048 here.

    float* out = (float*)d_out;
    const size_t OUT_ELEMS = (size_t)B_SZ * K_SEL * H_SZ;   // 16,777,216
    float* sel_f   = out + OUT_ELEMS;                        // [4,2048]
    float* targets = sel_f + (size_t)B_SZ * K_SEL;           // [4,4096]

    float* probs = (float*)d_ws;                             // 16384 f32
    int*   sel_i = (int*)(probs + (size_t)B_SZ * S_SZ);      // 8192 i32
    float* w_top = (float*)(sel_i + (size_t)B_SZ * K_SEL);   // 8192 f32

    // 16384 tokens / (16 tokens/wave * 8 waves/block) = 128 blocks
    mor_router_wmma<<<dim3(128), dim3(256), 0, stream>>>(x, w, probs);
    mor_select<<<dim3(B_SZ), dim3(1024), 0, stream>>>(probs, sel_f, targets, sel_i, w_top);
    mor_gather<<<dim3(B_SZ * K_SEL), dim3(256), 0, stream>>>(x, sel_i, w_top, out);
}